// RoIPointPool3d_55344948576550
// MI455X (gfx1250) — compile-verified
//
#include <hip/hip_runtime.h>
#include <hip/hip_bf16.h>

// RoIPointPool3d for MI455X (gfx1250, wave32).
//
// Sizes (fixed by the reference): B=4, N=16384, C=128, M=128, S=512, extra=1.0
//
// Roofline: output = 4*128*512*131 floats = 137 MB -> HBM-write bound,
// ~6-8 us floor at 23.3 TB/s. Points+features (34 MB) are L2-resident (192 MB).
// The 8.4M point-in-box tests are done with V_WMMA_F32_16X16X4_F32, three
// matmuls per 16-point x 16-box tile:
//   Dx = [px py 1 pz] x [ cos  sin  tx 0 ]^T   (local_x)
//   Dy = [px py 1 pz] x [-sin  cos  ty 0 ]^T   (local_y)
//   Dz = [px py 1 pz] x [  0    0  -zc 1 ]^T   (pz - z_center)
// All three land in the same D accumulator layout, so the in-box predicate is
// three register compares with no cross-lane traffic in the hot loop.

#define BB 4
#define NN 16384
#define CC 128
#define MM 128
#define SS 512

typedef __attribute__((ext_vector_type(2))) float v2f;
typedef __attribute__((ext_vector_type(8))) float v8f;

// ---------------------------------------------------------------------------
// Kernel 1: WMMA box-membership mask.
// grid.x = B*8 (one block per batch x 16-box group), block = 256 (8 waves).
// Each wave owns the same 16 boxes (columns) and strides over 16-point tiles.
// Output: per (bg, tile, boxInGroup) a 16-bit mask of which of the tile's 16
// points fall inside that box.  maskbuf layout: [bg][tile][box] u16.
// ---------------------------------------------------------------------------
__global__ __launch_bounds__(256) void roipool_mask_kernel(
    const float* __restrict__ points,   // (B, N, 3)
    const float* __restrict__ boxes,    // (B, M, 7)
    unsigned short* __restrict__ maskbuf)
{
  const int tid  = threadIdx.x;
  const int lane = tid & 31;
  const int wave = tid >> 5;            // 0..7
  const int bg   = blockIdx.x;          // 0..31  (b*8 + g)
  const int b    = bg >> 3;
  const int g    = bg & 7;
  const int n    = lane & 15;           // box column within group
  const bool lo  = (lane < 16);

  // --- per-lane box coefficients (both 16-lane halves compute the same box n)
  const float* bx7 = boxes + (size_t)(b * MM + g * 16 + n) * 7;
  const float cx = bx7[0], cy = bx7[1], cz = bx7[2];
  const float dxx = bx7[3], dyy = bx7[4], dzz = bx7[5], yaw = bx7[6];
  const float cosa = __cosf(yaw), sina = __sinf(yaw);
  // enlarged box: dx+=1, dy+=1, dz+=1, cz-=0.5 -> z-center = cz + dz/2
  const float hx = (dxx + 1.0f) * 0.5f;
  const float hy = (dyy + 1.0f) * 0.5f;
  const float hz = (dzz + 1.0f) * 0.5f;
  const float zc = cz + dzz * 0.5f;
  const float tx = -(cx * cosa + cy * sina);   // translation term for local_x
  const float ty =  (cx * sina - cy * cosa);   // translation term for local_y

  // B-matrix (4x16 f32): vgpr0 holds rows K0 (lanes 0-15) / K2 (lanes 16-31),
  // vgpr1 holds rows K1 / K3.  Column n = coefficients of box n.
  v2f Bx, By, Bz;
  Bx.x = lo ?  cosa : tx;   Bx.y = lo ? sina : 0.0f;   // local_x coeffs
  By.x = lo ? -sina : ty;   By.y = lo ? cosa : 0.0f;   // local_y coeffs
  Bz.x = lo ?  0.0f : -zc;  Bz.y = lo ? 0.0f : 1.0f;   // pz - zc coeffs

  unsigned short* mout = maskbuf + (size_t)bg * 1024 * 16;

  for (int tile = wave; tile < NN / 16; tile += 8) {
    // A-matrix (16x4 f32): lanes 0-15 hold point m=lane: vgpr0=K0(px),
    // vgpr1=K1(py); lanes 16-31 hold point m=lane-16: vgpr0=K2(=1),
    // vgpr1=K3(=pz).  Each lane loads the point whose K-slots it supplies.
    const int pt = tile * 16 + n;
    const float* p3 = points + ((size_t)b * NN + pt) * 3;
    const float pxv = p3[0], pyv = p3[1], pzv = p3[2];
    v2f A;
    A.x = lo ? pxv : 1.0f;
    A.y = lo ? pyv : pzv;

    v8f c0 = {};
    // D[m][n] layout: vgpr r, lane l -> m = r + 8*(l>=16), n = l&15.
    v8f Dx = __builtin_amdgcn_wmma_f32_16x16x4_f32(
        false, A, false, Bx, (short)0, c0, false, false);
    v8f Dy = __builtin_amdgcn_wmma_f32_16x16x4_f32(
        false, A, false, By, (short)0, c0, false, false);
    v8f Dz = __builtin_amdgcn_wmma_f32_16x16x4_f32(
        false, A, false, Bz, (short)0, c0, false, false);

    unsigned W[8];
#pragma unroll
    for (int r = 0; r < 8; ++r) {
      const bool pred = (fabsf(Dx[r]) < hx) &
                        (fabsf(Dy[r]) < hy) &
                        (fabsf(Dz[r]) <= hz);
      W[r] = (unsigned)__ballot(pred);              // uniform across wave
    }

    // Transpose ballots -> per-box 16-bit point mask (lanes 0-15 only).
    if (lo) {
      unsigned maskv = 0;
#pragma unroll
      for (int mrow = 0; mrow < 8; ++mrow)
        maskv |= ((W[mrow] >> n) & 1u) << mrow;            // points 0..7
#pragma unroll
      for (int mrow = 8; mrow < 16; ++mrow)
        maskv |= ((W[mrow - 8] >> (16 + n)) & 1u) << mrow; // points 8..15
      mout[(size_t)tile * 16 + n] = (unsigned short)maskv;
    }
  }
}

// ---------------------------------------------------------------------------
// Kernel 2: stable compaction.  One wave per box; lane order == tile order ==
// point order, so writes are in ascending point index (matches stable argsort).
// ---------------------------------------------------------------------------
__global__ __launch_bounds__(32) void roipool_compact_kernel(
    const unsigned short* __restrict__ maskbuf,
    int* __restrict__ idxbuf,       // (B*M, S)
    int* __restrict__ cntbuf)       // (B*M)
{
  const int boxId = blockIdx.x;           // 0..511
  const int b = boxId >> 7;
  const int m = boxId & 127;
  const int g = m >> 4;
  const int n = m & 15;
  const int lane = threadIdx.x;

  const unsigned short* mb =
      maskbuf + ((size_t)(b * 8 + g) * 1024) * 16 + n;
  int* myIdx = idxbuf + (size_t)boxId * SS;

  int running = 0;
  for (int tb = 0; tb < NN / 16; tb += 32) {
    const int tile = tb + lane;
    const unsigned maskv = mb[(size_t)tile * 16];
    const int c = __popc(maskv);

    // inclusive wave32 prefix sum
    int incl = c;
#pragma unroll
    for (int off = 1; off < 32; off <<= 1) {
      const int t = __shfl_up(incl, off);
      if (lane >= off) incl += t;
    }
    const int excl = incl - c;
    const int waveTotal = __shfl(incl, 31);

    int base = running + excl;
    unsigned mm2 = maskv;
    int j = 0;
    while (mm2) {
      const int bit = __ffs(mm2) - 1;
      const int pos = base + j;
      if (pos < SS) myIdx[pos] = tile * 16 + bit;
      mm2 &= mm2 - 1;
      ++j;
    }
    running += waveTotal;
  }
  if (lane == 0) cntbuf[boxId] = running;   // exact total count
}

// ---------------------------------------------------------------------------
// Kernel 3: gather.  One 128-thread block per (box, sample) row of 131 floats.
// Output layout: pooled_features (B,M,S,131) flat, then empty flags (B*M).
// ---------------------------------------------------------------------------
__global__ __launch_bounds__(128) void roipool_gather_kernel(
    const float* __restrict__ points,   // (B, N, 3)
    const float* __restrict__ feats,    // (B, N, 128)
    const int* __restrict__ idxbuf,
    const int* __restrict__ cntbuf,
    float* __restrict__ out)
{
  const size_t FEAT_TOTAL = (size_t)BB * MM * SS * (3 + CC);
  const int bs = blockIdx.x;            // 0 .. B*M*S-1
  const int s = bs & (SS - 1);
  const int boxId = bs >> 9;
  const int t = threadIdx.x;            // 0..127
  const int cnt = cntbuf[boxId];
  const size_t row = (size_t)bs * (3 + CC);

  if (cnt == 0) {
    out[row + t] = 0.0f;
    if (t < 3) out[row + 128 + t] = 0.0f;
    if (s == 0 && t == 0) out[FEAT_TOTAL + boxId] = 1.0f;
    return;
  }

  const int r = (s < cnt) ? s : (s % cnt);
  const int pidx = idxbuf[(size_t)boxId * SS + r];
  const int b = boxId >> 7;
  const float* p3 = points + ((size_t)b * NN + pidx) * 3;
  const float* f = feats + ((size_t)b * NN + pidx) * CC;

  out[row + t] = (t < 3) ? p3[t] : f[t - 3];        // elements 0..127
  if (t < 3) out[row + 128 + t] = f[125 + t];       // elements 128..130
  if (s == 0 && t == 0) out[FEAT_TOTAL + boxId] = 0.0f;
}

// ---------------------------------------------------------------------------
extern "C" void kernel_launch(void* const* d_in, const int* in_sizes, int n_in,
                              void* d_out, int out_size, void* d_ws, size_t ws_size,
                              hipStream_t stream) {
  (void)in_sizes; (void)n_in; (void)out_size; (void)ws_size;
  const float* points = (const float*)d_in[0];   // (4,16384,3)
  const float* feats  = (const float*)d_in[1];   // (4,16384,128)
  const float* boxes  = (const float*)d_in[2];   // (4,128,7)
  float* out = (float*)d_out;

  // workspace layout: [0,1MB) masks u16; [1MB,2MB) idxbuf i32; [2MB,+2KB) cnt
  unsigned short* maskbuf = (unsigned short*)d_ws;
  int* idxbuf = (int*)((char*)d_ws + (1u << 20));
  int* cntbuf = (int*)((char*)d_ws + (2u << 20));

  roipool_mask_kernel<<<BB * 8, 256, 0, stream>>>(points, boxes, maskbuf);
  roipool_compact_kernel<<<BB * MM, 32, 0, stream>>>(maskbuf, idxbuf, cntbuf);
  roipool_gather_kernel<<<BB * MM * SS, 128, 0, stream>>>(points, feats, idxbuf,
                                                          cntbuf, out);
}